// DecodeDetections_2671469658698
// MI455X (gfx1250) — compile-verified
//
#include <hip/hip_runtime.h>
#include <hip/hip_bf16.h>
#include <math.h>

// ---------------- problem constants ----------------
#define BB    8
#define NN    24564
#define NCH   93          // C + 12
#define NCLS  80          // foreground classes (C-1)
#define KK    400         // nms_max_output_size
#define TOPK  200
#define CONF_T 0.01f
#define IOU_T  0.45f
#define IMG_H 512.0f
#define IMG_W 512.0f

typedef __attribute__((ext_vector_type(2))) float v2f;
typedef __attribute__((ext_vector_type(8))) float v8f;
typedef __attribute__((ext_vector_type(4))) int   v4i;

#define AS1 __attribute__((address_space(1)))
#define AS3 __attribute__((address_space(3)))

// Async global->LDS copy path (CDNA5 GLOBAL_LOAD_ASYNC_TO_LDS_*, ASYNCcnt).
// Guarded so compilation never breaks if the toolchain lacks the builtins.
#if defined(__has_builtin)
#if __has_builtin(__builtin_amdgcn_global_load_async_to_lds_b128) && \
    __has_builtin(__builtin_amdgcn_s_wait_asynccnt)
#define HAVE_ASYNC_LDS 1
#endif
#endif

// workspace layout (floats):
//   boxes       : BB*NN*4               = 786048     @ 0
//   cand_scores : BB*NCLS*KK            = 256000     @ 786048
//   cand_boxes  : BB*NCLS*KK*4          = 1024000    @ 1042048
//   rows        : BB*NCLS*KK*6          = 1536000    @ 2066048
// total = 3602048 floats = ~14.4 MB

// ---------------------------------------------------------------------------
// Kernel A: decode anchors -> corner boxes.  Pure streaming, HBM bound.
// ---------------------------------------------------------------------------
__global__ void decode_boxes_kernel(const float* __restrict__ y,
                                    float* __restrict__ boxes) {
  int i = blockIdx.x * blockDim.x + threadIdx.x;   // box id over B*N
  if (i >= BB * NN) return;
  const float* p = y + (size_t)i * NCH + (NCH - 12);
  float ocx = p[0], ocy = p[1], ow = p[2], oh = p[3];
  float acx = p[4], acy = p[5], aw = p[6], ah = p[7];
  float vcx = p[8], vcy = p[9], vw = p[10], vh = p[11];
  float cx = ocx * vcx * aw + acx;
  float cy = ocy * vcy * ah + acy;
  float w  = __expf(ow * vw) * aw;
  float h  = __expf(oh * vh) * ah;
  float4 bb;
  bb.x = (cx - 0.5f * w) * IMG_W;
  bb.y = (cy - 0.5f * h) * IMG_H;
  bb.z = (cx + 0.5f * w) * IMG_W;
  bb.w = (cy + 0.5f * h) * IMG_H;
  ((float4*)boxes)[i] = bb;
}

// ---------------------------------------------------------------------------
// LDS bitonic primitives on packed (scoreBits<<32 | index) keys, descending.
// ---------------------------------------------------------------------------
__device__ __forceinline__ void bitonic_sort_desc(unsigned long long* a, int n) {
  for (int k = 2; k <= n; k <<= 1) {
    for (int j = k >> 1; j > 0; j >>= 1) {
      __syncthreads();
      for (int i = threadIdx.x; i < n; i += blockDim.x) {
        int ixj = i ^ j;
        if (ixj > i) {
          unsigned long long x = a[i], y = a[ixj];
          bool firstHalf = ((i & k) == 0);          // descending overall
          bool doSwap = firstHalf ? (x < y) : (x > y);
          if (doSwap) { a[i] = y; a[ixj] = x; }
        }
      }
    }
  }
  __syncthreads();
}

// Input must be a bitonic sequence; output is sorted descending.
// Only log2(n) stages instead of a full sort.
__device__ __forceinline__ void bitonic_merge_desc(unsigned long long* a, int n) {
  for (int j = n >> 1; j > 0; j >>= 1) {
    __syncthreads();
    for (int i = threadIdx.x; i < n; i += blockDim.x) {
      int ixj = i ^ j;
      if (ixj > i) {
        unsigned long long x = a[i], y = a[ixj];
        if (x < y) { a[i] = y; a[ixj] = x; }        // keep larger at low index
      }
    }
  }
  __syncthreads();
}

// ---------------------------------------------------------------------------
// Kernel B: per (batch, class) top-400 by score, sorted descending.
// Tournament top-list: sorted top-512 in LDS; per chunk: full sort of the
// chunk (45 stages) + element-wise max-combine (bitonic) + 9-stage merge.
// One block per (b, cls): 640 blocks x 256 threads (8 waves).
// ---------------------------------------------------------------------------
__global__ void topk_class_kernel(const float* __restrict__ y,
                                  const float* __restrict__ boxes,
                                  float* __restrict__ cand_scores,
                                  float* __restrict__ cand_boxes) {
  __shared__ unsigned long long s_top[512];
  __shared__ unsigned long long s_chunk[512];

  int bc  = blockIdx.x;            // 0..639
  int b   = bc / NCLS;
  int cls = bc - b * NCLS;         // 0..79
  int ch  = cls + 1;               // skip background channel 0

  for (int e = threadIdx.x; e < 512; e += blockDim.x) s_top[e] = 0ull;

  const int nchunks = (NN + 511) / 512;  // 48
  for (int cs = 0; cs < nchunks; ++cs) {
    for (int e = threadIdx.x; e < 512; e += blockDim.x) {
      int n = cs * 512 + e;
      float s = 0.0f;
      if (n < NN) s = y[((size_t)b * NN + n) * NCH + ch];
      s_chunk[e] = (((unsigned long long)__float_as_uint(s)) << 32) |
                   (unsigned)(n < NN ? n : 0);
    }
    bitonic_sort_desc(s_chunk, 512);
    // tournament: top[i] = max(top[i], chunk[511-i]) keeps the global top-512
    // and leaves s_top bitonic -> cheap merge instead of a full re-sort.
    for (int e = threadIdx.x; e < 512; e += blockDim.x) {
      unsigned long long a = s_top[e];
      unsigned long long c = s_chunk[511 - e];
      s_top[e] = (a < c) ? c : a;
    }
    bitonic_merge_desc(s_top, 512);  // leading barrier orders combine writes
  }

  // emit top-400 candidates (scores + gathered boxes)
  for (int k = threadIdx.x; k < KK; k += blockDim.x) {
    unsigned long long pk = s_top[k];
    unsigned idx = (unsigned)(pk & 0xFFFFFFFFull);
    if (idx >= NN) idx = NN - 1;
    float sc = __uint_as_float((unsigned)(pk >> 32));
    cand_scores[(size_t)bc * KK + k] = sc;
    ((float4*)cand_boxes)[(size_t)bc * KK + k] =
        ((const float4*)boxes)[(size_t)b * NN + idx];
  }
}

// ---------------------------------------------------------------------------
// Kernel C: greedy NMS per (batch, class) + row materialization.
//
// Phase 0: stage candidate boxes into LDS via async global->LDS DMA when
//          available (ASYNCcnt path), else plain loads.
// Phase 1 (parallel): build 400x400 suppression bitmask in LDS, tiled 16x16
//          per wave.  union = area_i + area_j - inter is ONE
//          V_WMMA_F32_16X16X4_F32 per tile (rank-2 outer product, C=-inter);
//          IoU test is the divide-free  inter > IOU_T * union.
// Phase 2 (wave 0): 400-step greedy loop reduced to 16-bit mask ANDs.
// ---------------------------------------------------------------------------
__global__ void nms_kernel(const float* __restrict__ cand_scores,
                           const float* __restrict__ cand_boxes,
                           float* __restrict__ rows) {
  __shared__ float4 s_bx[KK];                 // 6.4 KB
  __shared__ float  s_area[KK];               // 1.6 KB
  __shared__ float  s_sc[KK];                 // 1.6 KB
  __shared__ unsigned short s_mask[KK * 26];  // 400 rows x 25 tiles (+pad)
  __shared__ unsigned short s_kept[32];       // 400 keep bits

  int bc  = blockIdx.x;
  int b   = bc / NCLS;
  int cls = bc - b * NCLS;

#if defined(HAVE_ASYNC_LDS) && defined(__HIP_DEVICE_COMPILE__)
  // async DMA of the 6.4 KB box tile straight into LDS (no VGPR staging);
  // builtin prototype: (v4i addrspace(1)*, v4i addrspace(3)*, imm off, cpol)
  for (int j = threadIdx.x; j < KK; j += blockDim.x) {
    __builtin_amdgcn_global_load_async_to_lds_b128(
        (AS1 v4i*)(cand_boxes + ((size_t)bc * KK + j) * 4),
        (AS3 v4i*)&s_bx[j], 0, 0);
  }
  for (int j = threadIdx.x; j < KK; j += blockDim.x)
    s_sc[j] = cand_scores[(size_t)bc * KK + j];
  __builtin_amdgcn_s_wait_asynccnt(0);
  __syncthreads();
  for (int j = threadIdx.x; j < KK; j += blockDim.x) {
    float4 bb = s_bx[j];
    s_area[j] = fmaxf(bb.z - bb.x, 0.0f) * fmaxf(bb.w - bb.y, 0.0f);
  }
#else
  for (int j = threadIdx.x; j < KK; j += blockDim.x) {
    float4 bb = ((const float4*)cand_boxes)[(size_t)bc * KK + j];
    s_bx[j] = bb;
    s_area[j] = fmaxf(bb.z - bb.x, 0.0f) * fmaxf(bb.w - bb.y, 0.0f);
    s_sc[j] = cand_scores[(size_t)bc * KK + j];
  }
#endif
  __syncthreads();

  // initial keep = confidence threshold (25 x 16 bits)
  if (threadIdx.x < 25) {
    unsigned bits = 0;
    for (int q = 0; q < 16; ++q)
      bits |= (s_sc[threadIdx.x * 16 + q] > CONF_T ? 1u : 0u) << q;
    s_kept[threadIdx.x] = (unsigned short)bits;
  }

  // ---- Phase 1: suppression mask tiles (wave-uniform -> EXEC all ones) ----
  const int wv = threadIdx.x >> 5;
  const int lane = threadIdx.x & 31;
  const int nwaves = blockDim.x >> 5;
  for (int t = wv; t < 25 * 25; t += nwaves) {
    int ti = t / 25, tj = t - ti * 25;
    int col = tj * 16 + (lane & 15);
    float4 bj = s_bx[col];
    float  aj = s_area[col];

    // A (16x4): col0 = area_i, col1 = 1    | lanes 16..31 hold K=2,3 = 0
    v2f A;
    A.x = (lane < 16) ? s_area[ti * 16 + lane] : 0.0f;
    A.y = (lane < 16) ? 1.0f : 0.0f;
    // B (4x16): row0 = 1, row1 = area_j    | lanes 16..31 hold K=2,3 = 0
    v2f Bv;
    Bv.x = (lane < 16) ? 1.0f : 0.0f;
    Bv.y = (lane < 16) ? aj : 0.0f;

    float inter[8];
    v8f Cm;
#pragma unroll
    for (int r = 0; r < 8; ++r) {
      int m = ti * 16 + r + ((lane >> 4) << 3);   // C/D layout rows
      float4 bi = s_bx[m];
      float iw = fminf(bi.z, bj.z) - fmaxf(bi.x, bj.x);
      float ih = fminf(bi.w, bj.w) - fmaxf(bi.y, bj.y);
      float in_ = fmaxf(iw, 0.0f) * fmaxf(ih, 0.0f);
      inter[r] = in_;
      Cm[r] = -in_;
    }

    // union tile: D = area_i x 1 + 1 x area_j + (-inter)
    v8f U = __builtin_amdgcn_wmma_f32_16x16x4_f32(
        false, A, false, Bv, (short)0, Cm, false, false);

    // ballots first (cross-lane, all lanes active), stores in ONE branch
    unsigned bal[8];
#pragma unroll
    for (int r = 0; r < 8; ++r) {
      int row = ti * 16 + r + ((lane >> 4) << 3);
      bool p = (inter[r] > IOU_T * U[r]) && (col > row);
      bal[r] = (unsigned)__builtin_amdgcn_ballot_w32(p);
    }
    if (lane == 0) {
#pragma unroll
      for (int r = 0; r < 8; ++r) {
        // lanes 0..15 -> row ti*16+r ; lanes 16..31 -> row ti*16+r+8
        s_mask[(ti * 16 + r) * 26 + tj]     = (unsigned short)(bal[r] & 0xFFFFu);
        s_mask[(ti * 16 + r + 8) * 26 + tj] = (unsigned short)(bal[r] >> 16);
      }
    }
  }
  __syncthreads();

  // ---- Phase 2: serial greedy loop on wave 0 (bitwise ANDs only) ----
  if (threadIdx.x < 32) {
    volatile unsigned short* vk = s_kept;
    for (int i = 0; i < KK; ++i) {
      unsigned keep_i = (vk[i >> 4] >> (i & 15)) & 1u;
      if (keep_i) {
        int l = threadIdx.x;
        if (l < 25)
          vk[l] = (unsigned short)(vk[l] & ~s_mask[i * 26 + l]);
      }
    }
  }
  __syncthreads();

  // ---- materialize rows [class_id, score, x1, y1, x2, y2] (zero if culled) -
  for (int j = threadIdx.x; j < KK; j += blockDim.x) {
    bool keep = (s_kept[j >> 4] >> (j & 15)) & 1u;
    float4 bb = s_bx[j];
    size_t base = (((size_t)b * NCLS + cls) * KK + j) * 6;
    rows[base + 0] = keep ? (float)(cls + 1) : 0.0f;
    rows[base + 1] = keep ? s_sc[j] : 0.0f;
    rows[base + 2] = keep ? bb.x : 0.0f;
    rows[base + 3] = keep ? bb.y : 0.0f;
    rows[base + 4] = keep ? bb.z : 0.0f;
    rows[base + 5] = keep ? bb.w : 0.0f;
  }
}

// ---------------------------------------------------------------------------
// Kernel D: per-batch top-200 of 32000 rows by score, sorted descending.
// ---------------------------------------------------------------------------
__global__ void final_topk_kernel(const float* __restrict__ rows,
                                  float* __restrict__ out) {
  __shared__ unsigned long long s_top[256];
  __shared__ unsigned long long s_chunk[256];
  int b = blockIdx.x;

  for (int e = threadIdx.x; e < 256; e += blockDim.x) s_top[e] = 0ull;

  const int total = NCLS * KK;          // 32000
  const int nchunks = total / 256;      // 125
  for (int cs = 0; cs < nchunks; ++cs) {
    for (int e = threadIdx.x; e < 256; e += blockDim.x) {
      int idx = cs * 256 + e;
      float s = rows[((size_t)b * total + idx) * 6 + 1];
      s_chunk[e] = (((unsigned long long)__float_as_uint(s)) << 32) |
                   (unsigned)idx;
    }
    bitonic_sort_desc(s_chunk, 256);
    for (int e = threadIdx.x; e < 256; e += blockDim.x) {
      unsigned long long a = s_top[e];
      unsigned long long c = s_chunk[255 - e];
      s_top[e] = (a < c) ? c : a;
    }
    bitonic_merge_desc(s_top, 256);
  }

  if (threadIdx.x < TOPK) {
    unsigned idx = (unsigned)(s_top[threadIdx.x] & 0xFFFFFFFFull);
    size_t src = ((size_t)b * total + idx) * 6;
    size_t dst = ((size_t)b * TOPK + threadIdx.x) * 6;
    for (int q = 0; q < 6; ++q) out[dst + q] = rows[src + q];
  }
}

// ---------------------------------------------------------------------------
extern "C" void kernel_launch(void* const* d_in, const int* in_sizes, int n_in,
                              void* d_out, int out_size, void* d_ws, size_t ws_size,
                              hipStream_t stream) {
  const float* y = (const float*)d_in[0];
  float* ws = (float*)d_ws;

  float* boxes       = ws;                       // 786048 floats
  float* cand_scores = ws + 786048;              // 256000 floats
  float* cand_boxes  = ws + 1042048;             // 1024000 floats
  float* rows        = ws + 2066048;             // 1536000 floats

  int nboxes = BB * NN;
  decode_boxes_kernel<<<(nboxes + 255) / 256, 256, 0, stream>>>(y, boxes);
  topk_class_kernel<<<BB * NCLS, 256, 0, stream>>>(y, boxes, cand_scores,
                                                   cand_boxes);
  nms_kernel<<<BB * NCLS, 256, 0, stream>>>(cand_scores, cand_boxes, rows);
  final_topk_kernel<<<BB, 256, 0, stream>>>(rows, (float*)d_out);
}